// SSDLayer_21320217657904
// MI455X (gfx1250) — compile-verified
//
#include <hip/hip_runtime.h>
#include <stdint.h>

// SSD match + encode, fused, one workgroup per image.
// CDNA5 paths: TENSOR_LOAD_TO_LDS (TDM, TENSORcnt) for GT-box staging,
// GLOBAL_LOAD_ASYNC_TO_LDS_B32 (ASYNCcnt) for labels, global_prefetch,
// wave32 shuffle max-reductions, LDS-resident match state.

#define THREADS 256
#define MAXN    128          // max GT boxes per image (ref uses 100)
#define MAXP    8736         // max priors (ref uses 8732)
#define NWAVES  (THREADS / 32)
#define GT_PER_WAVE (MAXN / NWAVES)   // 16

typedef unsigned int uint4v __attribute__((ext_vector_type(4)));
typedef unsigned int uint8v __attribute__((ext_vector_type(8)));

__device__ __forceinline__ unsigned long long wave_max_u64(unsigned long long key) {
#pragma unroll
    for (int off = 16; off > 0; off >>= 1) {
        unsigned int hi  = (unsigned int)(key >> 32);
        unsigned int lo  = (unsigned int)key;
        unsigned int ohi = __shfl_xor(hi, off, 32);
        unsigned int olo = __shfl_xor(lo, off, 32);
        unsigned long long other = ((unsigned long long)ohi << 32) | (unsigned long long)olo;
        if (other > key) key = other;
    }
    return key;
}

__global__ __launch_bounds__(THREADS) void ssd_match_kernel(
    const float* __restrict__ gt_boxes,   // [B,N,4] xyxy
    const int*   __restrict__ gt_labels,  // [B,N]
    const float* __restrict__ priors,     // [P,4] cxcywh
    float* __restrict__ loc_out,          // [B,P,4]
    float* __restrict__ conf_out,         // [B,P] (conf written as float)
    int B, int N, int P)
{
    __shared__ float4         s_gt[MAXN];        // gt boxes xyxy
    __shared__ float          s_area[MAXN];      // gt areas
    __shared__ int            s_lab[MAXN];       // gt labels
    __shared__ unsigned int   s_bestprior[MAXN]; // argmax_p iou[n][p]
    __shared__ float          s_bov[MAXP];       // max_n iou[n][p]
    __shared__ unsigned short s_bidx[MAXP];      // argmax_n iou[n][p]

    const int b   = blockIdx.x;
    const int tid = threadIdx.x;

    // ---- Phase 0a: TDM copies this image's GT boxes (contiguous 16*N bytes)
    //      global -> LDS with a single 1-row tensor descriptor (TENSORcnt). ----
    if (tid < 32) {   // wave 0 issues the DMA (TDM ignores EXEC; issue is per wave)
        unsigned long long ga = (unsigned long long)(uintptr_t)(gt_boxes + (size_t)b * N * 4);
        unsigned int lds = (unsigned int)(uintptr_t)(&s_gt[0]);  // low 32b = LDS offset
        unsigned int nd0 = (unsigned int)(N * 4);                // elements (f32) in the row
        // D# group 0: count=1 | lds_addr | global_addr | type=2 (ISA 8.3)
        uint4v g0 = { 1u,
                      lds,
                      (unsigned int)ga,
                      (unsigned int)((ga >> 32) & 0x01FFFFFFu) | 0x80000000u };
        // D# group 1: data_size=4B; tensor_dim0=tile_dim0=nd0; dim0_stride=nd0 (ISA 8.4)
        uint8v g1 = { 0x00020000u,                       // wg_mask=0, data_size=2(4B)
                      (nd0 & 0xFFFFu) << 16,             // tensor_dim0[15:0]
                      (nd0 >> 16) | (1u << 16),          // tensor_dim0[31:16], tensor_dim1=1
                      (nd0 & 0xFFFFu) << 16,             // tile_dim0
                      0u,                                // tile_dim1=0 (unused), tile_dim2=0
                      nd0,                               // tensor_dim0_stride[31:0]
                      0u, 0u };
        asm volatile("tensor_load_to_lds %0, %1" :: "s"(g0), "s"(g1) : "memory");
    }

    // ---- Phase 0b: labels via async global->LDS (ASYNCcnt path) ----
    if (tid < N) {
        const int* lsrc = gt_labels + (size_t)b * N + tid;
        unsigned int llab = (unsigned int)(uintptr_t)(&s_lab[tid]);
        asm volatile("global_load_async_to_lds_b32 %0, %1, off"
                     :: "v"(llab), "v"(lsrc) : "memory");
    }
    // Warm the prior stream for this thread while the DMAs fly.
    __builtin_prefetch(priors + (size_t)tid * 4, 0, 1);

    asm volatile("s_wait_asynccnt 0x0" ::: "memory");
    if (tid < 32) __builtin_amdgcn_s_wait_tensorcnt(0);
    __syncthreads();

    if (tid < N) {
        float4 g = s_gt[tid];
        s_area[tid] = (g.z - g.x) * (g.w - g.y);
    }
    __syncthreads();

    const float4* pr4 = (const float4*)priors;

    // ---- Phase 2a: per-prior best GT (max/argmax over N, first-index ties) ----
    for (int p = tid; p < P; p += THREADS) {
        float4 pr = pr4[p];
        float px1 = pr.x - pr.z * 0.5f;
        float py1 = pr.y - pr.w * 0.5f;
        float px2 = pr.x + pr.z * 0.5f;
        float py2 = pr.y + pr.w * 0.5f;
        float pa  = (px2 - px1) * (py2 - py1);
        float bov = -1.0f;
        int   bidx = 0;
        for (int n = 0; n < N; ++n) {
            float4 g  = s_gt[n];
            float ix1 = fmaxf(g.x, px1), iy1 = fmaxf(g.y, py1);
            float ix2 = fminf(g.z, px2), iy2 = fminf(g.w, py2);
            float iw  = fmaxf(ix2 - ix1, 0.0f);
            float ih  = fmaxf(iy2 - iy1, 0.0f);
            float inter = iw * ih;
            float ov = inter / (s_area[n] + pa - inter + 1e-6f);
            if (ov > bov) { bov = ov; bidx = n; }
        }
        s_bov[p]  = bov;
        s_bidx[p] = (unsigned short)bidx;
    }

    // ---- Phase 2b: per-GT best prior. Each wave owns GTs n = wave + j*NWAVES,
    //      keys in registers, single coalesced pass over priors, wave32 reduce. ----
    {
        const int lane = tid & 31;
        const int wave = tid >> 5;
        unsigned long long key[GT_PER_WAVE];
#pragma unroll
        for (int j = 0; j < GT_PER_WAVE; ++j) key[j] = 0xFFFFFFFFull; // ov=0, p=0

        for (int p = lane; p < P; p += 32) {
            float4 pr = pr4[p];
            float px1 = pr.x - pr.z * 0.5f;
            float py1 = pr.y - pr.w * 0.5f;
            float px2 = pr.x + pr.z * 0.5f;
            float py2 = pr.y + pr.w * 0.5f;
            float pa  = (px2 - px1) * (py2 - py1);
            unsigned int lowbits = (unsigned int)(~p); // larger low word => smaller p (first-index tie)
            int n = wave;
#pragma unroll
            for (int j = 0; j < GT_PER_WAVE; ++j, n += NWAVES) {
                if (n < N) {
                    float4 g  = s_gt[n];
                    float ix1 = fmaxf(g.x, px1), iy1 = fmaxf(g.y, py1);
                    float ix2 = fminf(g.z, px2), iy2 = fminf(g.w, py2);
                    float iw  = fmaxf(ix2 - ix1, 0.0f);
                    float ih  = fmaxf(iy2 - iy1, 0.0f);
                    float inter = iw * ih;
                    float ov = inter / (s_area[n] + pa - inter + 1e-6f);
                    unsigned long long k =
                        ((unsigned long long)__float_as_uint(ov) << 32) | lowbits;
                    if (k > key[j]) key[j] = k;
                }
            }
        }
        int n = wave;
#pragma unroll
        for (int j = 0; j < GT_PER_WAVE; ++j, n += NWAVES) {
            if (n < N) {
                unsigned long long k = wave_max_u64(key[j]);
                if (lane == 0) s_bestprior[n] = ~(unsigned int)k; // decode prior idx
            }
        }
    }
    __syncthreads();

    // ---- Phase 3: force-match (ascending n, last writer wins), encode, store ----
    for (int p = tid; p < P; p += THREADS) {
        int forced = -1;
        for (int n = 0; n < N; ++n)
            if (s_bestprior[n] == (unsigned int)p) forced = n;

        float ov;
        int   idx;
        if (forced >= 0) { ov = 2.0f; idx = forced; }
        else             { ov = s_bov[p]; idx = (int)s_bidx[p]; }

        float4 g = s_gt[idx];
        int conf = (ov < 0.5f) ? 0 : (s_lab[idx] + 1);

        float4 pr = pr4[p];
        float mcx = (g.x + g.z) * 0.5f;
        float mcy = (g.y + g.w) * 0.5f;
        float mw  = fmaxf(g.z - g.x, 1e-6f);
        float mh  = fmaxf(g.w - g.y, 1e-6f);

        float4 o;
        o.x = (mcx - pr.x) / (0.1f * pr.z);
        o.y = (mcy - pr.y) / (0.1f * pr.w);
        o.z = logf(mw / pr.z) / 0.2f;
        o.w = logf(mh / pr.w) / 0.2f;

        ((float4*)loc_out)[(size_t)b * P + p] = o;
        conf_out[(size_t)b * P + p] = (float)conf;
    }
}

extern "C" void kernel_launch(void* const* d_in, const int* in_sizes, int n_in,
                              void* d_out, int out_size, void* d_ws, size_t ws_size,
                              hipStream_t stream) {
    (void)n_in; (void)d_ws; (void)ws_size;
    const float* gt_boxes  = (const float*)d_in[0];
    const int*   gt_labels = (const int*)  d_in[1];
    const float* priors    = (const float*)d_in[2];

    const int P = in_sizes[2] / 4;               // priors: [P,4]
    const int B = out_size / (5 * P);            // out = B*P*4 (loc) + B*P (conf)
    const int N = in_sizes[1] / B;               // labels: [B,N]

    float* loc  = (float*)d_out;
    float* conf = (float*)d_out + (size_t)B * P * 4;

    ssd_match_kernel<<<B, THREADS, 0, stream>>>(gt_boxes, gt_labels, priors,
                                                loc, conf, B, N, P);
}